// CapsuleLayer_19241453486473
// MI455X (gfx1250) — compile-verified
//
#include <hip/hip_runtime.h>
#include <math.h>

typedef __attribute__((ext_vector_type(2))) float v2f;
typedef __attribute__((ext_vector_type(8))) float v8f;

#define BB 64      // batch
#define II 2048    // input capsules
#define DD 8       // input capsule dim
#define OO 32      // output capsules
#define KK 16      // output capsule dim
#define BT 16      // b-tile rows (WMMA M)
#define NBT 4      // 64/16
#define NCHUNK 32  // i-chunks (grid.y)
#define ISTEP 64   // i per block
#define STAGE 32   // i per LDS staging round
#define NSTAGE 2   // ISTEP/STAGE
#define USTRIDE 20 // padded k-row stride for sh_u (80B, 16B-aligned)

// One routing iteration: recompute u_hat tiles with fp32 WMMA, do the
// agreement/softmax/weighted-sum locally per capsule i, emit per-chunk
// partial sums of s[b,o,k] (deterministic: no atomics).
__global__ __launch_bounds__(256) void caps_iter(
    const float* __restrict__ x, const float* __restrict__ W,
    const float* __restrict__ v0, const float* __restrict__ v1,
    float* __restrict__ s_partial, int t)
{
    __shared__ float sh_x[STAGE][BT][DD];      // 16 KB
    __shared__ float sh_u[OO][BT][USTRIDE];    // 40 KB

    const int tid   = threadIdx.x;
    const int wave  = tid >> 5;
    const int lane  = tid & 31;
    const int half  = lane >> 4;     // lane group (0: lanes 0-15, 1: 16-31)
    const int m     = lane & 15;     // row (A) / column (B,C) index
    const int btile = blockIdx.x;
    const int ichnk = blockIdx.y;
    const int ibase = ichnk * ISTEP;

    // ---- phase-2 thread mapping: lane = output capsule o, wave owns 2 b rows
    const int bl0 = 2 * wave, bl1 = bl0 + 1;
    const int bg0 = btile * BT + bl0, bg1 = bg0 + 1;
    const int oo  = lane;

    // previous v vectors kept in registers (tiny, L2-hot loads once)
    float vr0[2][KK], vr1[2][KK];
    if (t >= 1) {
        const float4* p0 = (const float4*)(v0 + ((size_t)bg0 * OO + oo) * KK);
        const float4* p1 = (const float4*)(v0 + ((size_t)bg1 * OO + oo) * KK);
#pragma unroll
        for (int q = 0; q < 4; ++q) {
            float4 a = p0[q], b = p1[q];
            vr0[0][4*q+0] = a.x; vr0[0][4*q+1] = a.y; vr0[0][4*q+2] = a.z; vr0[0][4*q+3] = a.w;
            vr0[1][4*q+0] = b.x; vr0[1][4*q+1] = b.y; vr0[1][4*q+2] = b.z; vr0[1][4*q+3] = b.w;
        }
    }
    if (t >= 2) {
        const float4* p0 = (const float4*)(v1 + ((size_t)bg0 * OO + oo) * KK);
        const float4* p1 = (const float4*)(v1 + ((size_t)bg1 * OO + oo) * KK);
#pragma unroll
        for (int q = 0; q < 4; ++q) {
            float4 a = p0[q], b = p1[q];
            vr1[0][4*q+0] = a.x; vr1[0][4*q+1] = a.y; vr1[0][4*q+2] = a.z; vr1[0][4*q+3] = a.w;
            vr1[1][4*q+0] = b.x; vr1[1][4*q+1] = b.y; vr1[1][4*q+2] = b.z; vr1[1][4*q+3] = b.w;
        }
    }

    float sacc[2][KK];
#pragma unroll
    for (int k = 0; k < KK; ++k) { sacc[0][k] = 0.f; sacc[1][k] = 0.f; }

    for (int stage = 0; stage < NSTAGE; ++stage) {
        const int i0 = ibase + stage * STAGE;
        // cooperative, coalesced load of x[b-tile, i-slice, :] into LDS
        for (int idx = tid; idx < BT * STAGE * DD; idx += 256) {
            int bl  = idx / (STAGE * DD);
            int rem = idx - bl * (STAGE * DD);
            int il  = rem / DD;
            int d   = rem - il * DD;
            sh_x[il][bl][d] =
                x[((size_t)(btile * BT + bl) * II + (i0 + il)) * DD + d];
        }
        __syncthreads();

        for (int il = 0; il < STAGE; ++il) {
            const int ig = i0 + il;

            // ===== phase 1: u_hat tile via V_WMMA_F32_16X16X4_F32 =====
            // A (16x4 f32): lanes 0-15 rows, VGPR0/1 = K0/K1 (half 0) or K2/K3 (half 1)
            v2f a_lo, a_hi;
            a_lo.x = sh_x[il][m][2 * half + 0];
            a_lo.y = sh_x[il][m][2 * half + 1];
            a_hi.x = sh_x[il][m][4 + 2 * half + 0];
            a_hi.y = sh_x[il][m][4 + 2 * half + 1];

#pragma unroll
            for (int j = 0; j < 4; ++j) {
                const int o = wave * 4 + j;
                const float* wp = W + (((size_t)ig * OO + o) * DD) * KK; // [d][k]
                v2f b_lo, b_hi;
                b_lo.x = wp[(2 * half + 0) * KK + m];
                b_lo.y = wp[(2 * half + 1) * KK + m];
                b_hi.x = wp[(4 + 2 * half + 0) * KK + m];
                b_hi.y = wp[(4 + 2 * half + 1) * KK + m];

                v8f u = {0.f, 0.f, 0.f, 0.f, 0.f, 0.f, 0.f, 0.f};
                u = __builtin_amdgcn_wmma_f32_16x16x4_f32(
                        false, a_lo, false, b_lo, (short)0, u, false, false);
                u = __builtin_amdgcn_wmma_f32_16x16x4_f32(
                        false, a_hi, false, b_hi, (short)0, u, false, false);

                // C layout: VGPR r, lanes0-15 -> M=r, lanes16-31 -> M=r+8, N=lane&15
#pragma unroll
                for (int r = 0; r < 8; ++r)
                    sh_u[o][r + 8 * half][m] = u[r];
            }
            __syncthreads();

            // ===== phase 2: agreements + softmax(o) + accumulate s =====
            float u0[KK], u1[KK];
            {
                const float4* r0 = (const float4*)&sh_u[oo][bl0][0];
                const float4* r1 = (const float4*)&sh_u[oo][bl1][0];
#pragma unroll
                for (int q = 0; q < 4; ++q) {
                    float4 a = r0[q], b = r1[q];
                    u0[4*q+0] = a.x; u0[4*q+1] = a.y; u0[4*q+2] = a.z; u0[4*q+3] = a.w;
                    u1[4*q+0] = b.x; u1[4*q+1] = b.y; u1[4*q+2] = b.z; u1[4*q+3] = b.w;
                }
            }

            float bb0 = 0.f, bb1 = 0.f;
            if (t >= 1) {
#pragma unroll
                for (int k = 0; k < KK; ++k) {
                    bb0 = fmaf(u0[k], vr0[0][k], bb0);
                    bb1 = fmaf(u1[k], vr0[1][k], bb1);
                }
            }
            if (t >= 2) {
#pragma unroll
                for (int k = 0; k < KK; ++k) {
                    bb0 = fmaf(u0[k], vr1[0][k], bb0);
                    bb1 = fmaf(u1[k], vr1[1][k], bb1);
                }
            }

            // softmax over the 32 output capsules = across the 32 lanes
            float mx0 = bb0, mx1 = bb1;
#pragma unroll
            for (int s = 16; s > 0; s >>= 1) {
                mx0 = fmaxf(mx0, __shfl_xor(mx0, s, 32));
                mx1 = fmaxf(mx1, __shfl_xor(mx1, s, 32));
            }
            float e0 = __expf(bb0 - mx0), e1 = __expf(bb1 - mx1);
            float sm0 = e0, sm1 = e1;
#pragma unroll
            for (int s = 16; s > 0; s >>= 1) {
                sm0 += __shfl_xor(sm0, s, 32);
                sm1 += __shfl_xor(sm1, s, 32);
            }
            const float c0 = e0 / sm0, c1 = e1 / sm1;

#pragma unroll
            for (int k = 0; k < KK; ++k) {
                sacc[0][k] = fmaf(c0, u0[k], sacc[0][k]);
                sacc[1][k] = fmaf(c1, u1[k], sacc[1][k]);
            }
            __syncthreads();
        }
    }

    // write per-chunk partial s (each (chunk,b,o,k) owned by exactly one thread)
    float4* sp0 = (float4*)(s_partial + (((size_t)ichnk * BB + bg0) * OO + oo) * KK);
    float4* sp1 = (float4*)(s_partial + (((size_t)ichnk * BB + bg1) * OO + oo) * KK);
#pragma unroll
    for (int q = 0; q < 4; ++q) {
        sp0[q] = make_float4(sacc[0][4*q+0], sacc[0][4*q+1], sacc[0][4*q+2], sacc[0][4*q+3]);
        sp1[q] = make_float4(sacc[1][4*q+0], sacc[1][4*q+1], sacc[1][4*q+2], sacc[1][4*q+3]);
    }
}

// Fixed-order reduction over i-chunks + squash -> v (deterministic)
__global__ __launch_bounds__(256) void caps_reduce(
    const float* __restrict__ s_partial, float* __restrict__ vout)
{
    const int bo = blockIdx.x * 256 + threadIdx.x; // 0..2047 = b*32+o
    float acc[KK];
#pragma unroll
    for (int k = 0; k < KK; ++k) acc[k] = 0.f;
    for (int ic = 0; ic < NCHUNK; ++ic) {
        const float4* p =
            (const float4*)(s_partial + ((size_t)ic * BB * OO + bo) * KK);
#pragma unroll
        for (int q = 0; q < 4; ++q) {
            float4 a = p[q];
            acc[4*q+0] += a.x; acc[4*q+1] += a.y;
            acc[4*q+2] += a.z; acc[4*q+3] += a.w;
        }
    }
    float s2 = 0.f;
#pragma unroll
    for (int k = 0; k < KK; ++k) s2 = fmaf(acc[k], acc[k], s2);
    const float scale = s2 / ((1.f + s2) * sqrtf(s2 + 1e-7f));
    float4* out = (float4*)(vout + (size_t)bo * KK);
#pragma unroll
    for (int q = 0; q < 4; ++q)
        out[q] = make_float4(scale * acc[4*q+0], scale * acc[4*q+1],
                             scale * acc[4*q+2], scale * acc[4*q+3]);
}

extern "C" void kernel_launch(void* const* d_in, const int* in_sizes, int n_in,
                              void* d_out, int out_size, void* d_ws, size_t ws_size,
                              hipStream_t stream)
{
    (void)in_sizes; (void)n_in; (void)out_size; (void)ws_size;
    const float* x = (const float*)d_in[0];   // [64,2048,8]
    const float* W = (const float*)d_in[1];   // [2048,32,8,16]
    float* out = (float*)d_out;               // [64,32,16]

    float* v0 = (float*)d_ws;                           // 32768 floats
    float* v1 = v0 + (size_t)BB * OO * KK;              // 32768 floats
    float* sp = v1 + (size_t)BB * OO * KK;              // 32*64*32*16 floats (4 MB)

    dim3 grid(NBT, NCHUNK);
    // iteration 0: uniform c = softmax(0)
    caps_iter<<<grid, 256, 0, stream>>>(x, W, nullptr, nullptr, sp, 0);
    caps_reduce<<<8, 256, 0, stream>>>(sp, v0);
    // iteration 1: agreements vs v0
    caps_iter<<<grid, 256, 0, stream>>>(x, W, v0, nullptr, sp, 1);
    caps_reduce<<<8, 256, 0, stream>>>(sp, v1);
    // iteration 2: agreements vs v0 + v1 (recomputed, no stored logits)
    caps_iter<<<grid, 256, 0, stream>>>(x, W, v0, v1, sp, 2);
    caps_reduce<<<8, 256, 0, stream>>>(sp, out);
}